// NGCF_63591285785041
// MI455X (gfx1250) — compile-verified
//
#include <hip/hip_runtime.h>
#include <math.h>

#define N_AUTHORS 50000
#define N_PAPERS  100000
#define N_TOT     (N_AUTHORS + N_PAPERS)
#define DIM       128
#define L_LAYERS  3
#define NEG_SLOPE 0.01f
#define EPS_NORM  1e-12f
#define W_PITCH   132   // 128 floats + 4 padding DWORDs per row (TDM pad) -> conflict-free LDS

typedef float v2f __attribute__((ext_vector_type(2)));
typedef float v8f __attribute__((ext_vector_type(8)));
typedef unsigned int u32x4 __attribute__((ext_vector_type(4)));
typedef int i32x4 __attribute__((ext_vector_type(4)));
typedef int i32x8 __attribute__((ext_vector_type(8)));

// Non-returning global f32 atomic add (L2-side RMW, STOREcnt-tracked).
__device__ __forceinline__ void atomic_add_f32_g(float* p, float v) {
  asm volatile("global_atomic_add_f32 %0, %1, off" :: "v"(p), "v"(v) : "memory");
}

// --- Tensor DMA descriptor builders (CDNA5 ISA) -------
// Group 0: [1:0]=count=1 | [63:32]=lds_addr | [120:64]=global_addr | [127:126]=type=2
__device__ __forceinline__ u32x4 tdm_g0(unsigned lds_addr, const void* gptr) {
  unsigned long long ga = (unsigned long long)(uintptr_t)gptr;
  u32x4 g;
  g[0] = 1u;                                                // count=1, user mode
  g[1] = lds_addr;                                          // LDS byte address
  g[2] = (unsigned)ga;                                      // global_addr[31:0]
  g[3] = (unsigned)((ga >> 32) & 0x1FFFFFFu) | (2u << 30);  // addr[56:32] | type=2
  return g;
}
// Group 1 for a 128x128 f32 tile with LDS padding 4 DW per 128 DW (pitch 132):
//  [17:16]=data_size=2 (4B) | [20]=pad_enable | [24:22]=pad_interval=6 (128 DW)
//  [31:25]=pad_amount=3 (4 DW) | tensor_dim0=128 | tensor_dim1=128
//  tile_dim0=128 | tile_dim1=128 | tensor_dim0_stride=128
__device__ __forceinline__ i32x8 tdm_g1_w128() {
  i32x8 g;
  g[0] = (int)((2u << 16) | (1u << 20) | (6u << 22) | (3u << 25));
  g[1] = (int)(128u << 16);  // tensor_dim0[15:0] in bits [31:16]
  g[2] = (int)(128u << 16);  // tensor_dim0 hi = 0 | tensor_dim1[15:0]
  g[3] = (int)(128u << 16);  // tensor_dim1 hi = 0 | tile_dim0 = 128
  g[4] = (int)(128u);        // tile_dim1 = 128, tile_dim2 = 0
  g[5] = (int)(128u);        // tensor_dim0_stride low 32 = 128
  g[6] = 0;
  g[7] = 0;
  return g;
}

// ---------------------------------------------------------------------------
// ego0 = concat(author, paper); also write out[:, 0:128]
// ---------------------------------------------------------------------------
__global__ void k_init(const float* __restrict__ auth, const float* __restrict__ pap,
                       float* __restrict__ ego, float* __restrict__ out) {
  long long i = (long long)blockIdx.x * blockDim.x + threadIdx.x;
  if (i >= (long long)N_TOT * (DIM / 4)) return;
  int row = (int)(i >> 5);           // DIM/4 == 32
  int q   = (int)(i & 31);
  float4 v = (row < N_AUTHORS)
                 ? ((const float4*)auth)[(size_t)row * (DIM / 4) + q]
                 : ((const float4*)pap)[(size_t)(row - N_AUTHORS) * (DIM / 4) + q];
  ((float4*)ego)[i] = v;
  ((float4*)(out + (size_t)row * 512))[q] = v;
}

// ---------------------------------------------------------------------------
// SpMM: side[r,:] += val * ego[c,:]   (one wave32 per edge, float4 per lane)
// ---------------------------------------------------------------------------
__global__ void k_spmm(const int* __restrict__ rows, const int* __restrict__ cols,
                       const float* __restrict__ vals, const float* __restrict__ ego,
                       float* __restrict__ side, int E) {
  int e = blockIdx.x * (blockDim.x >> 5) + (threadIdx.x >> 5);
  if (e >= E) return;
  int lane = threadIdx.x & 31;
  int r = rows[e];
  int c = cols[e];
  float v = vals[e];
  float4 x = ((const float4*)(ego + (size_t)c * DIM))[lane];
  float* dst = side + (size_t)r * DIM + lane * 4;
  atomic_add_f32_g(dst + 0, v * x.x);
  atomic_add_f32_g(dst + 1, v * x.y);
  atomic_add_f32_g(dst + 2, v * x.z);
  atomic_add_f32_g(dst + 3, v * x.w);
}

// ---------------------------------------------------------------------------
// Fused: ego_new = leaky(side @ W1^T + b1) + leaky((ego*side) @ W2^T + b2)
// 256 threads = 8 waves; wave w -> 16-row block (blockIdx*8 + w).
// W1/W2 staged in LDS by the Tensor Data Mover (one tensor_load_to_lds each,
// issued by wave 0, s_wait_tensorcnt 0, then workgroup barrier). LDS pitch 132
// makes the per-K b64 B-fragment reads hit all 64 banks exactly once.
// V_WMMA_F32_16X16X4_F32, K swept 128/4; all 8 col tiles x 2 branches live
// in 128 accumulator VGPRs so side/ego rows stream exactly once.
// ---------------------------------------------------------------------------
__global__ void __launch_bounds__(256) k_gemm(
    const float* __restrict__ ego, const float* __restrict__ side,
    const float* __restrict__ W1, const float* __restrict__ b1,
    const float* __restrict__ W2, const float* __restrict__ b2,
    float* __restrict__ egoNew) {
  extern __shared__ float lds[];
  float* ldsW1 = lds;                       // 128 * 132 floats
  float* ldsW2 = lds + 128 * W_PITCH;       // 128 * 132 floats

  // --- TDM stage: W1, W2 -> LDS (wave 0 issues both DMAs) -------------------
  if (threadIdx.x < 32) {
    u32x4 g0a = tdm_g0((unsigned)(uintptr_t)ldsW1, W1);
    u32x4 g0b = tdm_g0((unsigned)(uintptr_t)ldsW2, W2);
    i32x8 g1 = tdm_g1_w128();
    i32x4 z4 = {0, 0, 0, 0};
    i32x8 z8 = {0, 0, 0, 0, 0, 0, 0, 0};
    __builtin_amdgcn_tensor_load_to_lds(g0a, g1, z4, z4, z8, 0);
    __builtin_amdgcn_tensor_load_to_lds(g0b, g1, z4, z4, z8, 0);
    __builtin_amdgcn_s_wait_tensorcnt(0);
  }
  __syncthreads();

  const int lane = threadIdx.x & 31;
  const int wave = threadIdx.x >> 5;
  const int rb = blockIdx.x * 8 + wave;
  if (rb < N_TOT / 16) {                    // wave-uniform: EXEC all-1s for WMMA
    const int row0  = rb * 16;
    const int m     = lane & 15;
    const int khalf = lane >> 4;
    const int kh2   = khalf * 2;

    v8f acc1[8], acc2[8];
#pragma unroll
    for (int t = 0; t < 8; ++t) {
      acc1[t] = (v8f){0.f, 0.f, 0.f, 0.f, 0.f, 0.f, 0.f, 0.f};
      acc2[t] = (v8f){0.f, 0.f, 0.f, 0.f, 0.f, 0.f, 0.f, 0.f};
    }

    const float* srow = side + (size_t)(row0 + m) * DIM + kh2;
    const float* erow = ego  + (size_t)(row0 + m) * DIM + kh2;
    const float* w1p  = ldsW1 + (size_t)m * W_PITCH + kh2;
    const float* w2p  = ldsW2 + (size_t)m * W_PITCH + kh2;

    for (int k0 = 0; k0 < DIM; k0 += 4) {
      v2f aS = *(const v2f*)(srow + k0);    // sum-branch A fragment
      v2f aE = *(const v2f*)(erow + k0);
      v2f aB = aS * aE;                     // bilinear-branch A fragment
#pragma unroll
      for (int t = 0; t < 8; ++t) {
        v2f bw1 = *(const v2f*)(w1p + (size_t)t * 16 * W_PITCH + k0);
        v2f bw2 = *(const v2f*)(w2p + (size_t)t * 16 * W_PITCH + k0);
        acc1[t] = __builtin_amdgcn_wmma_f32_16x16x4_f32(
            false, aS, false, bw1, (short)0, acc1[t], false, false);
        acc2[t] = __builtin_amdgcn_wmma_f32_16x16x4_f32(
            false, aB, false, bw2, (short)0, acc2[t], false, false);
      }
    }

    // Epilogue: +bias, LeakyReLU per branch, sum, store.
    // C/D layout: c[r] is (row = row0 + r + 8*khalf, col = t*16 + m)
#pragma unroll
    for (int t = 0; t < 8; ++t) {
      float bb1 = b1[t * 16 + m];
      float bb2 = b2[t * 16 + m];
#pragma unroll
      for (int r8 = 0; r8 < 8; ++r8) {
        float x1 = acc1[t][r8] + bb1;
        x1 = (x1 >= 0.f) ? x1 : NEG_SLOPE * x1;
        float x2 = acc2[t][r8] + bb2;
        x2 = (x2 >= 0.f) ? x2 : NEG_SLOPE * x2;
        egoNew[(size_t)(row0 + r8 + 8 * khalf) * DIM + t * 16 + m] = x1 + x2;
      }
    }
  }
}

// ---------------------------------------------------------------------------
// L2-normalize each row of x into out[:, colOff : colOff+128]
// ---------------------------------------------------------------------------
__global__ void k_norm(const float* __restrict__ x, float* __restrict__ out, int colOff) {
  int row = blockIdx.x * (blockDim.x >> 5) + (threadIdx.x >> 5);
  if (row >= N_TOT) return;
  int lane = threadIdx.x & 31;
  float4 v = ((const float4*)(x + (size_t)row * DIM))[lane];
  float ss = v.x * v.x + v.y * v.y + v.z * v.z + v.w * v.w;
#pragma unroll
  for (int off = 16; off > 0; off >>= 1) ss += __shfl_xor(ss, off, 32);
  float n = sqrtf(ss);
  float s = 1.f / fmaxf(n, EPS_NORM);
  float4 o = make_float4(v.x * s, v.y * s, v.z * s, v.w * s);
  ((float4*)(out + (size_t)row * 512 + colOff))[lane] = o;
}

// ---------------------------------------------------------------------------
extern "C" void kernel_launch(void* const* d_in, const int* in_sizes, int n_in,
                              void* d_out, int out_size, void* d_ws, size_t ws_size,
                              hipStream_t stream) {
  const float* auth = (const float*)d_in[0];
  const float* pap  = (const float*)d_in[1];
  const int*   rows = (const int*)d_in[2];
  const int*   cols = (const int*)d_in[3];
  const float* vals = (const float*)d_in[4];
  const float* W1   = (const float*)d_in[5];
  const float* b1   = (const float*)d_in[6];
  const float* W2   = (const float*)d_in[7];
  const float* b2   = (const float*)d_in[8];
  float* out = (float*)d_out;
  const int E = in_sizes[2];

  const size_t NF = (size_t)N_TOT * DIM;
  float* egoA = (float*)d_ws;     // 76.8 MB
  float* egoB = egoA + NF;        // 76.8 MB
  float* side = egoB + NF;        // 76.8 MB

  {
    long long tot = (long long)N_TOT * (DIM / 4);
    int bs = 256;
    unsigned g = (unsigned)((tot + bs - 1) / bs);
    k_init<<<g, bs, 0, stream>>>(auth, pap, egoA, out);
  }

  const size_t ldsBytes = 2u * 128u * W_PITCH * sizeof(float);  // 135168 B

  float* cur = egoA;
  float* nxt = egoB;
  for (int k = 0; k < L_LAYERS; ++k) {
    (void)hipMemsetAsync(side, 0, NF * sizeof(float), stream);
    {
      int wpb = 8;  // 256 threads = 8 wave32s = 8 edges/block
      unsigned g = (unsigned)((E + wpb - 1) / wpb);
      k_spmm<<<g, 256, 0, stream>>>(rows, cols, vals, cur, side, E);
    }
    {
      int rbs = N_TOT / 16;                  // 9375 row blocks
      unsigned g = (unsigned)((rbs + 7) / 8);
      k_gemm<<<g, 256, ldsBytes, stream>>>(cur, side,
                                           W1 + (size_t)k * DIM * DIM, b1 + (size_t)k * DIM,
                                           W2 + (size_t)k * DIM * DIM, b2 + (size_t)k * DIM,
                                           nxt);
    }
    {
      unsigned g = (unsigned)((N_TOT + 7) / 8);
      k_norm<<<g, 256, 0, stream>>>(nxt, out, DIM * (k + 1));
    }
    float* t = cur; cur = nxt; nxt = t;
  }
}